// Loss2_57200374448351
// MI455X (gfx1250) — compile-verified
//
#include <hip/hip_runtime.h>
#include <stdint.h>

// ---------------------------------------------------------------------------
// Loss = 0.5*BCEwithLogits(mean) + 1.0*Dice(global) + 0.5*Topo(sum per-sample)
//
// Topo trick: targets are binary -> sorted target is a step function with
// z = N-K zeros. mean|sort(p)-sort(t)| = (2*sum_of_z_smallest_p + K - S)/N.
// sum_of_z_smallest is computed with a 3-pass (11/11/10-bit) histogram radix
// select on the monotone bit-transform of the logits (sigmoid is monotone,
// so selecting on x == selecting on p), accumulating per-bin sigmoid sums.
//
// Roofline: ~256MB total traffic -> ~11us at 23.3TB/s HBM. x (64MB) is
// re-read by passes B/C, t (64MB) is read once: t is loaded with th:NT so
// x stays resident in the 192MB L2 and passes B/C run at L2 bandwidth.
// Pass1 uses the CDNA5 async global->LDS path (global_load_async_to_lds_b128
// + s_wait_asynccnt) with double buffering; each thread stages and consumes
// its own 16B so the pipeline needs no block barriers. Select passes use
// global_prefetch_b8. Sums use global_atomic_add_f64 for accuracy.
// ---------------------------------------------------------------------------

#define BATCH 64
#define NPS (512 * 512)                       // elements per sample
#define CHUNKS 32                             // blocks per sample
#define CHUNK_ELEMS (NPS / CHUNKS)            // 8192
#define THREADS 256
#define TILE 1024                             // elements per staged tile
#define TILES_PER_BLOCK (CHUNK_ELEMS / TILE)  // 8

struct Ws {
  double bce_sum;
  double ptsum;
  double topo_sum;
  double Ssamp[BATCH];
  double sumBefore[BATCH];
  unsigned Kcnt[BATCH];
  unsigned zarr[BATCH];
  unsigned selBits[BATCH];
  unsigned cntBefore[BATCH];
  unsigned hcntA[BATCH][2048];
  unsigned hcntB[BATCH][2048];
  unsigned hcntC[BATCH][1024];
  double hsumA[BATCH][2048];
  double hsumB[BATCH][2048];
  double hsumC[BATCH][1024];
};

// Order-preserving float -> uint key (ascending)
__device__ __forceinline__ unsigned f2u(float f) {
  unsigned u = __float_as_uint(f);
  return (u & 0x80000000u) ? ~u : (u | 0x80000000u);
}
__device__ __forceinline__ float u2f(unsigned k) {
  unsigned u = (k & 0x80000000u) ? (k & 0x7fffffffu) : ~k;
  return __uint_as_float(u);
}

__device__ __forceinline__ double wredd(double v) {
#pragma unroll
  for (int o = 16; o > 0; o >>= 1) v += __shfl_down(v, o, 32);
  return v;
}
__device__ __forceinline__ unsigned wredu(unsigned v) {
#pragma unroll
  for (int o = 16; o > 0; o >>= 1) v += __shfl_down(v, o, 32);
  return v;
}

// CDNA5 async global->LDS copy of 16B per lane (tracked by ASYNCcnt).
// Regular-temporal flavor for data that will be re-read (logits)...
__device__ __forceinline__ void async16(unsigned lds_addr, const float* gptr) {
  asm volatile("global_load_async_to_lds_b128 %0, %1, off"
               :
               : "v"(lds_addr), "v"((unsigned long long)(uintptr_t)gptr)
               : "memory");
}
// ...and non-temporal flavor for stream-once data (targets), so it does not
// evict the logits from L2 before the select passes re-read them.
__device__ __forceinline__ void async16_nt(unsigned lds_addr, const float* gptr) {
  asm volatile("global_load_async_to_lds_b128 %0, %1, off th:TH_LOAD_NT"
               :
               : "v"(lds_addr), "v"((unsigned long long)(uintptr_t)gptr)
               : "memory");
}

__global__ void k_init(unsigned long long* w, size_t n64) {
  size_t i = (size_t)blockIdx.x * blockDim.x + threadIdx.x;
  size_t st = (size_t)gridDim.x * blockDim.x;
  for (; i < n64; i += st) w[i] = 0ull;
}

// Fused pass: BCE sum, global p*t sum, per-sample K and S, 2048-bin top-level
// histogram (count + sigmoid-sum). Async double-buffered LDS staging.
__global__ __launch_bounds__(THREADS) void k_pass1(const float* __restrict__ x,
                                                   const float* __restrict__ t,
                                                   Ws* __restrict__ ws) {
  __shared__ float sx[2][TILE];
  __shared__ float st_[2][TILE];
  __shared__ unsigned hc[2048];
  __shared__ float hs[2048];

  const int s = blockIdx.y;
  const int tid = threadIdx.x;
  for (int i = tid; i < 2048; i += THREADS) { hc[i] = 0u; hs[i] = 0.0f; }
  __syncthreads();

  const size_t base = (size_t)s * NPS + (size_t)blockIdx.x * CHUNK_ELEMS;
  const float* gx = x + base;
  const float* gt = t + base;
  const int e0 = tid * 4;

  // stage tile 0
  async16((unsigned)(uintptr_t)&sx[0][e0], gx + e0);
  async16_nt((unsigned)(uintptr_t)&st_[0][e0], gt + e0);

  double bce = 0.0, ss = 0.0, spt = 0.0;
  unsigned kc = 0u;

  for (int tile = 0; tile < TILES_PER_BLOCK; ++tile) {
    const int buf = tile & 1;
    if (tile + 1 < TILES_PER_BLOCK) {
      async16((unsigned)(uintptr_t)&sx[buf ^ 1][e0], gx + (tile + 1) * TILE + e0);
      async16_nt((unsigned)(uintptr_t)&st_[buf ^ 1][e0], gt + (tile + 1) * TILE + e0);
      // 2 newer ops in flight; async loads complete in order -> tile `tile` done
      asm volatile("s_wait_asynccnt 2" ::: "memory");
    } else {
      asm volatile("s_wait_asynccnt 0" ::: "memory");
    }
    const float4 xv = *(const float4*)&sx[buf][e0];
    const float4 tv = *(const float4*)&st_[buf][e0];
    const float xa[4] = {xv.x, xv.y, xv.z, xv.w};
    const float ta[4] = {tv.x, tv.y, tv.z, tv.w};
#pragma unroll
    for (int j = 0; j < 4; ++j) {
      const float xi = xa[j], ti = ta[j];
      bce += (double)(fmaxf(xi, 0.0f) - xi * ti + log1pf(expf(-fabsf(xi))));
      const float p = 1.0f / (1.0f + expf(-xi));
      ss += (double)p;
      spt += (double)(p * ti);
      kc += (ti != 0.0f) ? 1u : 0u;
      const unsigned u = f2u(xi);
      const unsigned bin = u >> 21;
      atomicAdd(&hc[bin], 1u);
      atomicAdd(&hs[bin], p);
    }
  }

  bce = wredd(bce);
  ss = wredd(ss);
  spt = wredd(spt);
  const unsigned kr = wredu(kc);
  if ((tid & 31) == 0) {
    atomicAdd(&ws->bce_sum, bce);
    atomicAdd(&ws->ptsum, spt);
    atomicAdd(&ws->Ssamp[s], ss);
    atomicAdd(&ws->Kcnt[s], kr);
  }
  __syncthreads();
  for (int i = tid; i < 2048; i += THREADS) {
    const unsigned c = hc[i];
    if (c) {
      atomicAdd(&ws->hcntA[s][i], c);
      atomicAdd(&ws->hsumA[s][i], (double)hs[i]);
    }
  }
}

__global__ void k_scanA(Ws* __restrict__ ws) {
  const int s = blockIdx.x;
  if (threadIdx.x != 0) return;
  const unsigned z = (unsigned)NPS - ws->Kcnt[s];
  unsigned cum = 0, b0 = 2047;
  double sum = 0.0;
  for (unsigned b = 0; b < 2048; ++b) {
    const unsigned c = ws->hcntA[s][b];
    if (cum + c >= z) { b0 = b; break; }
    cum += c;
    sum += ws->hsumA[s][b];
  }
  ws->zarr[s] = z;
  ws->selBits[s] = b0;
  ws->cntBefore[s] = cum;
  ws->sumBefore[s] = sum;
}

__global__ __launch_bounds__(THREADS) void k_passB(const float* __restrict__ x,
                                                   Ws* __restrict__ ws) {
  __shared__ unsigned hc[2048];
  __shared__ float hs[2048];
  const int s = blockIdx.y;
  const int tid = threadIdx.x;
  const unsigned b0 = ws->selBits[s];
  for (int i = tid; i < 2048; i += THREADS) { hc[i] = 0u; hs[i] = 0.0f; }
  __syncthreads();
  const float* gx = x + (size_t)s * NPS + (size_t)blockIdx.x * CHUNK_ELEMS;
  for (int i = tid * 4; i < CHUNK_ELEMS; i += THREADS * 4) {
    __builtin_prefetch(gx + i + THREADS * 4, 0, 0);  // global_prefetch_b8
    const float4 v = *(const float4*)(gx + i);
    const float a[4] = {v.x, v.y, v.z, v.w};
#pragma unroll
    for (int j = 0; j < 4; ++j) {
      const unsigned u = f2u(a[j]);
      if ((u >> 21) == b0) {
        const unsigned bin = (u >> 10) & 2047u;
        atomicAdd(&hc[bin], 1u);
        atomicAdd(&hs[bin], 1.0f / (1.0f + expf(-a[j])));
      }
    }
  }
  __syncthreads();
  for (int i = tid; i < 2048; i += THREADS) {
    const unsigned c = hc[i];
    if (c) {
      atomicAdd(&ws->hcntB[s][i], c);
      atomicAdd(&ws->hsumB[s][i], (double)hs[i]);
    }
  }
}

__global__ void k_scanB(Ws* __restrict__ ws) {
  const int s = blockIdx.x;
  if (threadIdx.x != 0) return;
  const unsigned z = ws->zarr[s];
  unsigned cb = ws->cntBefore[s], b1 = 2047;
  double sb = ws->sumBefore[s];
  for (unsigned b = 0; b < 2048; ++b) {
    const unsigned c = ws->hcntB[s][b];
    if (cb + c >= z) { b1 = b; break; }
    cb += c;
    sb += ws->hsumB[s][b];
  }
  ws->selBits[s] = (ws->selBits[s] << 11) | b1;  // top 22 bits
  ws->cntBefore[s] = cb;
  ws->sumBefore[s] = sb;
}

__global__ __launch_bounds__(THREADS) void k_passC(const float* __restrict__ x,
                                                   Ws* __restrict__ ws) {
  __shared__ unsigned hc[1024];
  __shared__ float hs[1024];
  const int s = blockIdx.y;
  const int tid = threadIdx.x;
  const unsigned sel22 = ws->selBits[s];
  for (int i = tid; i < 1024; i += THREADS) { hc[i] = 0u; hs[i] = 0.0f; }
  __syncthreads();
  const float* gx = x + (size_t)s * NPS + (size_t)blockIdx.x * CHUNK_ELEMS;
  for (int i = tid * 4; i < CHUNK_ELEMS; i += THREADS * 4) {
    __builtin_prefetch(gx + i + THREADS * 4, 0, 0);
    const float4 v = *(const float4*)(gx + i);
    const float a[4] = {v.x, v.y, v.z, v.w};
#pragma unroll
    for (int j = 0; j < 4; ++j) {
      const unsigned u = f2u(a[j]);
      if ((u >> 10) == sel22) {
        const unsigned bin = u & 1023u;
        atomicAdd(&hc[bin], 1u);
        atomicAdd(&hs[bin], 1.0f / (1.0f + expf(-a[j])));
      }
    }
  }
  __syncthreads();
  for (int i = tid; i < 1024; i += THREADS) {
    const unsigned c = hc[i];
    if (c) {
      atomicAdd(&ws->hcntC[s][i], c);
      atomicAdd(&ws->hsumC[s][i], (double)hs[i]);
    }
  }
}

__global__ void k_scanC(Ws* __restrict__ ws) {
  const int s = blockIdx.x;
  if (threadIdx.x != 0) return;
  const unsigned z = ws->zarr[s];
  unsigned cb = ws->cntBefore[s], b2 = 1023;
  double sb = ws->sumBefore[s];
  for (unsigned b = 0; b < 1024; ++b) {
    const unsigned c = ws->hcntC[s][b];
    if (cb + c >= z) { b2 = b; break; }
    cb += c;
    sb += ws->hsumC[s][b];
  }
  const unsigned r = z - cb;  // ties at the threshold value
  double sum_small = sb;
  if (r) {  // guard: when r==0 tau may decode to garbage/NaN; 0*NaN would poison
    const unsigned tau = (ws->selBits[s] << 10) | b2;
    const float xt = u2f(tau);
    const double ptau = 1.0 / (1.0 + exp(-(double)xt));
    sum_small += (double)r * ptau;
  }
  const double K = (double)ws->Kcnt[s];
  const double topo = (2.0 * sum_small + K - ws->Ssamp[s]) / (double)NPS;
  atomicAdd(&ws->topo_sum, topo);
}

__global__ void k_final(const Ws* __restrict__ ws, float* __restrict__ out) {
  if (threadIdx.x != 0 || blockIdx.x != 0) return;
  double T = 0.0, S = 0.0;
  for (int i = 0; i < BATCH; ++i) {
    T += (double)ws->Kcnt[i];
    S += ws->Ssamp[i];
  }
  const double tot = (double)BATCH * (double)NPS;
  const double bce = ws->bce_sum / tot;
  const double dice = 1.0 - (2.0 * ws->ptsum + 1e-6) / (S + T + 1e-6);
  out[0] = (float)(0.5 * bce + 1.0 * dice + 0.5 * ws->topo_sum);
}

extern "C" void kernel_launch(void* const* d_in, const int* in_sizes, int n_in,
                              void* d_out, int out_size, void* d_ws, size_t ws_size,
                              hipStream_t stream) {
  (void)in_sizes; (void)n_in; (void)out_size; (void)ws_size;
  const float* x = (const float*)d_in[0];
  const float* t = (const float*)d_in[1];
  Ws* ws = (Ws*)d_ws;
  float* out = (float*)d_out;

  const size_t n64 = sizeof(Ws) / 8;
  k_init<<<2048, 256, 0, stream>>>((unsigned long long*)d_ws, n64);

  dim3 grid(CHUNKS, BATCH);
  k_pass1<<<grid, THREADS, 0, stream>>>(x, t, ws);
  k_scanA<<<BATCH, 32, 0, stream>>>(ws);
  k_passB<<<grid, THREADS, 0, stream>>>(x, ws);
  k_scanB<<<BATCH, 32, 0, stream>>>(ws);
  k_passC<<<grid, THREADS, 0, stream>>>(x, ws);
  k_scanC<<<BATCH, 32, 0, stream>>>(ws);
  k_final<<<1, 1, 0, stream>>>(ws, out);
}